// GCNWithDropout_1030792151719
// MI455X (gfx1250) — compile-verified
//
#include <hip/hip_runtime.h>

typedef __attribute__((ext_vector_type(2))) float v2f;
typedef __attribute__((ext_vector_type(8))) float v8f;

#define HID 64   // hidden dim (fixed by the model)

// ---------------------------------------------------------------- degrees ---
__global__ void k_deg_init(float* __restrict__ deg, int n) {
  int i = blockIdx.x * blockDim.x + threadIdx.x;
  if (i < n) deg[i] = 1.0f;                 // self-loop contributes 1
}

__global__ void k_deg_count(const long long* __restrict__ dst,
                            float* __restrict__ deg, int e) {
  int i = blockIdx.x * blockDim.x + threadIdx.x;
  if (i < e) atomicAdd(&deg[(int)dst[i]], 1.0f);   // global_atomic_add_f32 in L2
}

__global__ void k_rsqrt(float* __restrict__ d, int n) {
  int i = blockIdx.x * blockDim.x + threadIdx.x;
  if (i < n) d[i] = rsqrtf(d[i]);           // deg >= 1 always
}

// ------------------------------------------------------- H = X @ W^T (WMMA) --
// One wave computes a 16-row x 64-col slab of H using V_WMMA_F32_16X16X4_F32.
// A (16x4 f32): lanes 0-15 hold {K=kk,kk+1} of row M=lane; lanes 16-31 hold
// {K=kk+2,kk+3} of row M=lane-16.  B (4x16): mirrored, B[k][n] = W[n][k].
// C/D (16x16 f32, 8 VGPRs): lane n = l%16, vgpr r -> row r + 8*(l/16).
template <int RELU>
__global__ void k_gemm_wmma(const float* __restrict__ X, const float* __restrict__ W,
                            float* __restrict__ Hm, int n) {
  const int lane = threadIdx.x & 31;
  const int wave = threadIdx.x >> 5;
  const int rt   = blockIdx.x * 8 + wave;          // 16-row tile index
  const int ntiles = (n + 15) >> 4;
  if (rt >= ntiles) return;                        // whole-wave exit: EXEC all-1s at WMMA
  const int half = lane >> 4;                      // 0 or 1 (K sub-pair)
  const int r16  = lane & 15;

  int rowA = rt * 16 + r16;
  if (rowA > n - 1) rowA = n - 1;                  // clamp loads on tail tile

  v8f acc0 = {}, acc1 = {}, acc2 = {}, acc3 = {};

#pragma unroll
  for (int kk = 0; kk < HID; kk += 4) {
    const int ko = kk + 2 * half;
    v2f a = *(const v2f*)(X + (size_t)rowA * HID + ko);
    if (RELU) { a.x = fmaxf(a.x, 0.0f); a.y = fmaxf(a.y, 0.0f); }
    v2f b0 = *(const v2f*)(W + (size_t)(r16 +  0) * HID + ko);
    v2f b1 = *(const v2f*)(W + (size_t)(r16 + 16) * HID + ko);
    v2f b2 = *(const v2f*)(W + (size_t)(r16 + 32) * HID + ko);
    v2f b3 = *(const v2f*)(W + (size_t)(r16 + 48) * HID + ko);
    acc0 = __builtin_amdgcn_wmma_f32_16x16x4_f32(false, a, false, b0, (short)0, acc0, false, false);
    acc1 = __builtin_amdgcn_wmma_f32_16x16x4_f32(false, a, false, b1, (short)0, acc1, false, false);
    acc2 = __builtin_amdgcn_wmma_f32_16x16x4_f32(false, a, false, b2, (short)0, acc2, false, false);
    acc3 = __builtin_amdgcn_wmma_f32_16x16x4_f32(false, a, false, b3, (short)0, acc3, false, false);
  }

#pragma unroll
  for (int r = 0; r < 8; ++r) {
    const int row = rt * 16 + r + 8 * half;
    if (row < n) {
      float* o = Hm + (size_t)row * HID + r16;
      o[0]  = acc0[r];
      o[16] = acc1[r];
      o[32] = acc2[r];
      o[48] = acc3[r];
    }
  }
}

// ----------------------------------------- agg = h * dinv^2 (self-loop) + b --
__global__ void k_init_agg(const float* __restrict__ Hm, const float* __restrict__ dinv,
                           const float* __restrict__ bias, float* __restrict__ AGG, int nh) {
  int gid = blockIdx.x * blockDim.x + threadIdx.x;
  if (gid >= nh) return;
  int v = gid >> 6;
  int j = gid & (HID - 1);
  float di = dinv[v];
  AGG[gid] = Hm[gid] * di * di + bias[j];
}

// ----------------------- agg[dst] += h[src] * dinv[src]*dinv[dst] (scatter) --
__global__ void k_scatter(const long long* __restrict__ src, const long long* __restrict__ dst,
                          const float* __restrict__ dinv, const float* __restrict__ Hm,
                          float* __restrict__ AGG, int e) {
  int gid = blockIdx.x * blockDim.x + threadIdx.x;   // e*64 = 64M < 2^31
  if (gid >= e * HID) return;
  int j  = gid & (HID - 1);
  int ed = gid >> 6;
  int s = (int)src[ed];
  int d = (int)dst[ed];
  float nrm = dinv[s] * dinv[d];
  atomicAdd(&AGG[(size_t)d * HID + j], Hm[(size_t)s * HID + j] * nrm);
}

// ------------------------------- out[e] = <z[s], z[d]> (wave-per-edge, w32) --
__global__ void k_decode(const long long* __restrict__ si, const long long* __restrict__ di,
                         const float* __restrict__ Z, float* __restrict__ out, int l) {
  int wid  = (blockIdx.x * blockDim.x + threadIdx.x) >> 5;
  int lane = threadIdx.x & 31;
  if (wid >= l) return;
  int s = (int)si[wid];
  int d = (int)di[wid];
  const float* zs = Z + (size_t)s * HID;
  const float* zd = Z + (size_t)d * HID;
  float p = zs[lane] * zd[lane] + zs[lane + 32] * zd[lane + 32];
#pragma unroll
  for (int m = 16; m >= 1; m >>= 1) p += __shfl_xor(p, m, 32);
  if (lane == 0) out[wid] = p;
}

// ---------------------------------------------------------------- launcher ---
extern "C" void kernel_launch(void* const* d_in, const int* in_sizes, int n_in,
                              void* d_out, int out_size, void* d_ws, size_t ws_size,
                              hipStream_t stream) {
  (void)n_in; (void)out_size; (void)ws_size;
  const long long* ei  = (const long long*)d_in[0];   // [2, E] int64
  const long long* eli = (const long long*)d_in[1];   // [2, L] int64
  const float* emb = (const float*)d_in[2];           // [N, 64]
  const float* W1  = (const float*)d_in[3];
  const float* b1  = (const float*)d_in[4];
  const float* W2  = (const float*)d_in[5];
  const float* b2  = (const float*)d_in[6];
  const int E = in_sizes[0] / 2;
  const int L = in_sizes[1] / 2;
  const int N = in_sizes[2] / HID;
  float* out = (float*)d_out;

  // Workspace: dinv[N] | bufA[N*64] | bufB[N*64]  (~52 MB, L2-resident)
  char* ws = (char*)d_ws;
  float* dinv = (float*)ws;
  size_t o1 = ((size_t)N * sizeof(float) + 255) & ~(size_t)255;
  float* bufA = (float*)(ws + o1);
  size_t o2 = o1 + ((((size_t)N * HID * sizeof(float)) + 255) & ~(size_t)255);
  float* bufB = (float*)(ws + o2);

  const int T = 256;                                   // 8 wave32 per block
  const int gN  = (N + T - 1) / T;
  const int gE  = (E + T - 1) / T;
  const int nh  = N * HID;
  const int gNH = (nh + T - 1) / T;
  const int gEH = (E * HID + T - 1) / T;
  const int ntiles  = (N + 15) / 16;
  const int gGemm   = (ntiles + 7) / 8;
  const int gDec    = (L * 32 + T - 1) / T;

  // symmetric-normalization coefficients
  k_deg_init <<<gN, T, 0, stream>>>(dinv, N);
  k_deg_count<<<gE, T, 0, stream>>>(ei + E, dinv, E);
  k_rsqrt    <<<gN, T, 0, stream>>>(dinv, N);

  // layer 1: h = emb @ W1^T ; agg = scatter(h) + self-loop + b1
  k_gemm_wmma<0><<<gGemm, T, 0, stream>>>(emb, W1, bufA, N);
  k_init_agg   <<<gNH, T, 0, stream>>>(bufA, dinv, b1, bufB, nh);
  k_scatter    <<<gEH, T, 0, stream>>>(ei, ei + E, dinv, bufA, bufB, E);

  // layer 2: h = relu(agg) @ W2^T (ReLU fused into A-load) ; z -> bufB
  k_gemm_wmma<1><<<gGemm, T, 0, stream>>>(bufB, W2, bufA, N);
  k_init_agg   <<<gNH, T, 0, stream>>>(bufA, dinv, b2, bufB, nh);
  k_scatter    <<<gEH, T, 0, stream>>>(ei, ei + E, dinv, bufA, bufB, E);

  // decode: per-label-edge dot product of endpoint embeddings
  k_decode<<<gDec, T, 0, stream>>>(eli, eli + L, bufB, out, L);
}